// GateBasedQuantumLSTMCell_21182778704493
// MI455X (gfx1250) — compile-verified
//
#include <hip/hip_runtime.h>

typedef __attribute__((ext_vector_type(2))) float v2f;
typedef __attribute__((ext_vector_type(8))) float v8f;

#define DIM 4096

// Composition of the 11-CNOT entangle ladder as an index permutation.
// Reference: state_out[i] = state_in[f_phase1(f_phase2(i))], each CNOT: t ^= c.
// wire w lives at bit (11-w); phase2 ctrls = wires 1,3,5,7,9 (bits 10,8,6,4,2),
// phase1 ctrls = wires 0,2,4,6,8,10 (bits 11,9,7,5,3,1); target bit = ctrl bit - 1.
__device__ __forceinline__ int eperm(int i) {
  i ^= (i & 0x554) >> 1;   // phase-2 CNOTs
  i ^= (i & 0xAAA) >> 1;   // phase-1 CNOTs
  return i;
}

// Build 16x16 group unitary U = kron(RY(a0),RY(a1),RY(a2),RY(a3)), row-major.
// RY = [[c,-s],[s,c]], wire (base+k) acts on bit (3-k) of the 4-bit group index.
__device__ __forceinline__ void build_u(float* U, const float* ang4, int tid) {
  const int m = tid >> 4, n = tid & 15;
  float v = 1.f;
  #pragma unroll
  for (int k = 0; k < 4; ++k) {
    float s, c;
    __sincosf(0.5f * ang4[k], &s, &c);
    const int rm = (m >> (3 - k)) & 1;
    const int rn = (n >> (3 - k)) & 1;
    v *= (rm == rn) ? c : (rm ? s : -s);
  }
  U[tid] = v;
}

// Apply U along one 16-wide axis of S[a][b][c] (flat 4096 floats), src -> dst.
// MODE 2: axis a (wires 0-3, stride 256)   out = U @ T      (16 tiles over b)
// MODE 1: axis b (wires 4-7, stride 16)    out = U @ S_a    (16 tiles over a)
// MODE 0: axis c (wires 8-11, stride 1)    out = tile @ U^T (16 row-tiles)
// Each tile is one 16x16x16 matmul = 4 x V_WMMA_F32_16X16X4_F32.
// WMMA f32 layouts (wave32): A 16x4: lane=M%16 + 16*(K>=2), vgpr=K%2;
// B 4x16: lane=N + 16*(K>=2)... vgpr=K%2; C/D: lane=N + 16*(M>=8), vgpr=M%8.
template <int MODE, bool PERM>
__device__ __forceinline__ void apply_group(const float* __restrict__ src,
                                            float* __restrict__ dst,
                                            const float* __restrict__ U,
                                            int tid) {
  const int lane = tid & 31;
  const int wid  = tid >> 5;        // 0..7 (8 waves / 256-thread block)
  const int lm   = lane & 15;
  const int half = lane >> 4;       // 0 or 1
  #pragma unroll
  for (int tt = 0; tt < 2; ++tt) {
    const int t = wid * 2 + tt;     // tile 0..15
    v8f c = {0.f, 0.f, 0.f, 0.f, 0.f, 0.f, 0.f, 0.f};
    #pragma unroll
    for (int j = 0; j < 4; ++j) {
      v2f a, b;
      #pragma unroll
      for (int i = 0; i < 2; ++i) {
        const int k = 4 * j + half * 2 + i;   // K index covered by this lane/vgpr
        if (MODE == 0) {
          int idx = (16 * t + lm) * 16 + k;   // A = state row-tile
          if (PERM) idx = eperm(idx);
          a[i] = src[idx];
          b[i] = U[lm * 16 + k];              // B = U^T : B[k][n] = U[n][k]
        } else if (MODE == 1) {
          a[i] = U[lm * 16 + k];              // A = U : A[m][k]
          int idx = t * 256 + k * 16 + lm;    // B = S_a rows (stride 16)
          if (PERM) idx = eperm(idx);
          b[i] = src[idx];
        } else {
          a[i] = U[lm * 16 + k];
          int idx = k * 256 + t * 16 + lm;    // B = T_b rows (stride 256)
          if (PERM) idx = eperm(idx);
          b[i] = src[idx];
        }
      }
      c = __builtin_amdgcn_wmma_f32_16x16x4_f32(false, a, false, b,
                                                (short)0, c, false, false);
    }
    #pragma unroll
    for (int v = 0; v < 8; ++v) {
      const int m = v + half * 8;
      int o;
      if (MODE == 0)      o = (16 * t + m) * 16 + lm;
      else if (MODE == 1) o = t * 256 + m * 16 + lm;
      else                o = m * 256 + t * 16 + lm;
      dst[o] = c[v];
    }
  }
}

// One block = one (gate, batch) statevector, fully LDS-resident (ping-pong).
__global__ __launch_bounds__(256)
void vqc_kernel(const float* __restrict__ x, const float* __restrict__ h_prev,
                const float* __restrict__ w_gates, float* __restrict__ gate_ws) {
  __shared__ float buf[2][DIM];     // 32 KB ping-pong state
  __shared__ float U[256];          // current 16x16 group unitary
  __shared__ float angs[5][12];     // step 0 = encode angles, 1..4 = layer weights

  const int tid  = threadIdx.x;
  const int gate = blockIdx.x >> 10;
  const int b    = blockIdx.x & 1023;

  if (tid < 12) angs[0][tid] = (tid < 4) ? x[b * 4 + tid] : h_prev[b * 8 + (tid - 4)];
  if (tid < 48) angs[1 + tid / 12][tid % 12] = w_gates[gate * 48 + tid];
  #pragma unroll
  for (int r = 0; r < 16; ++r) buf[0][r * 256 + tid] = 0.015625f;  // DIM^-0.5

  int cur = 0;
  for (int step = 0; step < 5; ++step) {
    const bool ent = (step > 0);
    #pragma unroll
    for (int g = 0; g < 3; ++g) {
      __syncthreads();                       // prior apply done reading U/buffers
      build_u(U, &angs[step][4 * g], tid);
      __syncthreads();                       // U + prior stores visible
      const float* src = buf[cur];
      float* dst = buf[cur ^ 1];
      if (g == 0) {                          // wires 0-3 (high); fold entangle perm
        if (ent) apply_group<2, true >(src, dst, U, tid);
        else     apply_group<2, false>(src, dst, U, tid);
      } else if (g == 1) {
        apply_group<1, false>(src, dst, U, tid);   // wires 4-7 (mid)
      } else {
        apply_group<0, false>(src, dst, U, tid);   // wires 8-11 (low)
      }
      cur ^= 1;
    }
  }
  __syncthreads();

  // <Z_w> for wires 0..7 : sum of |amp|^2 signed by bit (11-w)
  float acc[8] = {0, 0, 0, 0, 0, 0, 0, 0};
  const float* st = buf[cur];
  #pragma unroll
  for (int r = 0; r < 16; ++r) {
    const int i = r * 256 + tid;
    const float p = st[i] * st[i];
    #pragma unroll
    for (int w = 0; w < 8; ++w)
      acc[w] += ((i >> (11 - w)) & 1) ? -p : p;
  }
  float* red = buf[cur ^ 1];
  #pragma unroll
  for (int w = 0; w < 8; ++w) red[w * 256 + tid] = acc[w];
  __syncthreads();
  for (int s = 128; s > 0; s >>= 1) {
    if (tid < s) {
      #pragma unroll
      for (int w = 0; w < 8; ++w) red[w * 256 + tid] += red[w * 256 + tid + s];
    }
    __syncthreads();
  }
  if (tid < 8) gate_ws[(gate * 1024 + b) * 8 + tid] = red[tid * 256];
}

// LSTM combine + tiny (1024x8)@(8x16) output projection.
__global__ __launch_bounds__(256)
void lstm_out_kernel(const float* __restrict__ gate_ws,
                     const float* __restrict__ c_prev,
                     const float* __restrict__ w_out,
                     const float* __restrict__ b_out,
                     float* __restrict__ out) {
  const int bidx = blockIdx.x * 256 + threadIdx.x;
  if (bidx >= 1024) return;
  float ht[8];
  #pragma unroll
  for (int h = 0; h < 8; ++h) {
    const float iv = gate_ws[(0 * 1024 + bidx) * 8 + h];
    const float fv = gate_ws[(1 * 1024 + bidx) * 8 + h];
    const float gv = gate_ws[(2 * 1024 + bidx) * 8 + h];
    const float ov = gate_ws[(3 * 1024 + bidx) * 8 + h];
    const float ig = 1.f / (1.f + __expf(-iv));
    const float fg = 1.f / (1.f + __expf(-fv));
    const float gg = tanhf(gv);
    const float og = 1.f / (1.f + __expf(-ov));
    const float ct = fg * c_prev[bidx * 8 + h] + ig * gg;
    const float hv = og * tanhf(ct);
    ht[h] = hv;
    out[16384 + bidx * 8 + h] = hv;   // h_t
    out[24576 + bidx * 8 + h] = ct;   // c_t
  }
  #pragma unroll
  for (int o = 0; o < 16; ++o) {
    float s = b_out[o];
    #pragma unroll
    for (int h = 0; h < 8; ++h) s += ht[h] * w_out[o * 8 + h];
    out[bidx * 16 + o] = s;           // out
  }
}

extern "C" void kernel_launch(void* const* d_in, const int* in_sizes, int n_in,
                              void* d_out, int out_size, void* d_ws, size_t ws_size,
                              hipStream_t stream) {
  const float* x       = (const float*)d_in[0];  // (1024, 4)
  const float* h_prev  = (const float*)d_in[1];  // (1024, 8)
  const float* c_prev  = (const float*)d_in[2];  // (1024, 8)
  const float* w_gates = (const float*)d_in[3];  // (4, 4, 12)
  const float* w_out   = (const float*)d_in[4];  // (16, 8)
  const float* b_out   = (const float*)d_in[5];  // (16,)
  float* out = (float*)d_out;                    // out(16384) | h_t(8192) | c_t(8192)
  float* gate_ws = (float*)d_ws;                 // (4, 1024, 8) f32 = 128 KB

  vqc_kernel<<<4096, 256, 0, stream>>>(x, h_prev, w_gates, gate_ws);
  lstm_out_kernel<<<4, 256, 0, stream>>>(gate_ws, c_prev, w_out, b_out, out);
}